// Sampler_738734375009
// MI455X (gfx1250) — compile-verified
//
#include <hip/hip_runtime.h>
#include <hip/hip_bf16.h>
#include <math.h>

#define TILE    2048     // floats per staged tile (8 KB)
#define NBUF    8        // ring-buffer depth (2 async b128 per lane per tile)
#define BLOCK   256
#define SPLITS  2        // workgroups per row
#define CAP     512      // candidate slots per row in workspace

// CDNA5 async global -> LDS copy, 16 bytes per lane, tracked by ASYNCcnt.
__device__ __forceinline__ void async_load_b128(unsigned lds_addr, unsigned long long gaddr) {
  asm volatile("global_load_async_to_lds_b128 %0, %1, off"
               :: "v"(lds_addr), "v"(gaddr)
               : "memory");
}

// ---------------- K0: zero per-row candidate counters ----------------
__global__ void k0_init(unsigned* __restrict__ wsCnt, int B) {
  int i = blockIdx.x * blockDim.x + threadIdx.x;
  if (i < B) wsCnt[i] = 0u;
}

// ---------------- K1: per-half-row scan ----------------
// Sweep A (HBM, deep async ring): chunk max only.
// Threshold: tau = 63rd-largest per-thread chunk max  =>  >=63 elements >= tau,
// and every global top-63 element is >= tau of its own chunk.
// Sweep B (L2): softmax partial sums with fixed base M_wg + candidate compaction.
__global__ __launch_bounds__(BLOCK) void k1_scan(
    const float* __restrict__ logits,
    const float* __restrict__ temperature,
    unsigned* __restrict__ wsCnt,          // [B]
    float*    __restrict__ wsM,            // [B*SPLITS]
    float*    __restrict__ wsS1,           // [B*SPLITS]
    float*    __restrict__ wsSt,           // [B*SPLITS]
    float*    __restrict__ wsVal,          // [B*CAP]
    int*      __restrict__ wsIdx,          // [B*CAP]
    int V)
{
  __shared__ __align__(16) float stage[NBUF][TILE];   // 64 KB ring
  __shared__ float rA[BLOCK];
  __shared__ float rB[BLOCK];

  const int wg    = blockIdx.x;
  const int row   = wg / SPLITS;
  const int split = wg % SPLITS;
  const int tid   = threadIdx.x;

  const int Vh  = V / SPLITS;
  const int c0  = split * Vh;
  const int len = (split == SPLITS - 1) ? (V - c0) : Vh;
  const float* cp = logits + (size_t)row * V + c0;

  float temp  = temperature[row];
  float tsafe = (temp < 1e-5f) ? 1.0f : temp;
  float inv_t = 1.0f / tsafe;

  // ---- Sweep A: max only, 8-deep async ring (HBM pass) ----
  float m = -3.402823466e+38f;
  const int NT = len / TILE;
  unsigned ldsBase = (unsigned)(size_t)(void*)&stage[0][0];

  auto issue_tile = [&](int tt) {
    unsigned dst = ldsBase + (unsigned)((tt & (NBUF - 1)) * (TILE * 4)) + tid * 16u;
    const float* g = cp + (size_t)tt * TILE + tid * 4;
    async_load_b128(dst,                  (unsigned long long)(size_t)g);
    async_load_b128(dst + (TILE / 2) * 4, (unsigned long long)(size_t)(g + TILE / 2));
  };

  const int pro = (NT < NBUF - 1) ? NT : (NBUF - 1);
  for (int p = 0; p < pro; ++p) issue_tile(p);

  for (int t = 0; t < NT; ++t) {
    const int buf = t & (NBUF - 1);
    if (t + NBUF - 1 < NT) {
      issue_tile(t + NBUF - 1);
      // 8 tiles x 2 instrs outstanding; oldest tile complete at <= 14
      asm volatile("s_wait_asynccnt 14" ::: "memory");
    } else {
      asm volatile("s_wait_asynccnt 0" ::: "memory");
    }
    float4 a = *(const float4*)&stage[buf][tid * 4];
    float4 b = *(const float4*)&stage[buf][TILE / 2 + tid * 4];
    float ma = fmaxf(fmaxf(a.x, a.y), fmaxf(a.z, a.w));
    float mb = fmaxf(fmaxf(b.x, b.y), fmaxf(b.z, b.w));
    m = fmaxf(m, fmaxf(ma, mb));
    // ring wraps next iteration: LDS reads must retire before re-issue
    asm volatile("s_wait_dscnt 0" ::: "memory");
  }
  for (int i = NT * TILE + tid; i < len; i += BLOCK)
    m = fmaxf(m, cp[i]);

  // ---- Sort the 256 thread maxima (descending bitonic) ----
  rA[tid] = m;
  __syncthreads();
  for (int k = 2; k <= BLOCK; k <<= 1) {
    for (int j = k >> 1; j > 0; j >>= 1) {
      int e = tid, p = e ^ j;
      if (p > e) {
        float ve = rA[e], vp = rA[p];
        bool up = ((e & k) == 0);
        if (up ? (ve < vp) : (ve > vp)) { rA[e] = vp; rA[p] = ve; }
      }
      __syncthreads();
    }
  }
  const float Mwg = rA[0];
  const float tau = rA[62];
  __syncthreads();

  // ---- Sweep B (L2-resident): partial sums + candidate compaction ----
  float s1a = 0.f, s1b = 0.f, sta = 0.f, stb = 0.f;
  for (int t = 0; t < NT; ++t) {
    if (t + 2 < NT)   // pull next-next tile toward the WGP (global_prefetch_b8)
      __builtin_prefetch(cp + (size_t)(t + 2) * TILE + tid * 4, 0, 0);
    float4 a = *(const float4*)(cp + (size_t)t * TILE + tid * 4);
    float4 b = *(const float4*)(cp + (size_t)t * TILE + TILE / 2 + tid * 4);
    #pragma unroll
    for (int c = 0; c < 4; ++c) {
      float xa = (&a.x)[c], xb = (&b.x)[c];
      float da = xa - Mwg,  db = xb - Mwg;
      s1a += __expf(da);          s1b += __expf(db);
      sta += __expf(da * inv_t);  stb += __expf(db * inv_t);
      if (xa >= tau) {
        unsigned slot = atomicAdd(&wsCnt[row], 1u);
        if (slot < CAP) {
          wsVal[(size_t)row * CAP + slot] = xa;
          wsIdx[(size_t)row * CAP + slot] = c0 + t * TILE + tid * 4 + c;
        }
      }
      if (xb >= tau) {
        unsigned slot = atomicAdd(&wsCnt[row], 1u);
        if (slot < CAP) {
          wsVal[(size_t)row * CAP + slot] = xb;
          wsIdx[(size_t)row * CAP + slot] = c0 + t * TILE + TILE / 2 + tid * 4 + c;
        }
      }
    }
  }
  for (int i = NT * TILE + tid; i < len; i += BLOCK) {
    float x = cp[i];
    float d = x - Mwg;
    s1a += __expf(d);
    sta += __expf(d * inv_t);
    if (x >= tau) {
      unsigned slot = atomicAdd(&wsCnt[row], 1u);
      if (slot < CAP) {
        wsVal[(size_t)row * CAP + slot] = x;
        wsIdx[(size_t)row * CAP + slot] = c0 + i;
      }
    }
  }

  // ---- Reduce partial sums (fixed base => plain adds) ----
  rA[tid] = s1a + s1b; rB[tid] = sta + stb;
  __syncthreads();
  for (int off = BLOCK / 2; off > 0; off >>= 1) {
    if (tid < off) { rA[tid] += rA[tid + off]; rB[tid] += rB[tid + off]; }
    __syncthreads();
  }
  if (tid == 0) { wsM[wg] = Mwg; wsS1[wg] = rA[0]; wsSt[wg] = rB[0]; }
}

// ---------------- K2: merge partials, sort candidates, sample ----------------
__global__ __launch_bounds__(BLOCK) void k2_finalize(
    const float* __restrict__ temperature,
    const float* __restrict__ top_p,
    const int*   __restrict__ top_k,
    const float* __restrict__ gumbel,
    const unsigned* __restrict__ wsCnt,
    const float* __restrict__ wsM,
    const float* __restrict__ wsS1,
    const float* __restrict__ wsSt,
    const float* __restrict__ wsVal,
    const int*   __restrict__ wsIdx,
    int*   __restrict__ out_ids,
    int*   __restrict__ out_idx,
    float* __restrict__ out_lp,
    int*   __restrict__ out_rank,
    int V, int nl)
{
  __shared__ float candVal[CAP];
  __shared__ int   candIdx[CAP];
  __shared__ float pj[64], cumb[64], score[64];
  __shared__ float sM, sS1, sSt;

  const int row = blockIdx.x;
  const int tid = threadIdx.x;

  float temp   = temperature[row];
  bool  greedy = (temp < 1e-5f);
  float tsafe  = greedy ? 1.0f : temp;
  float inv_t  = 1.0f / tsafe;
  float topp   = top_p[row];
  int   kk     = top_k[row];
  kk = kk < 1 ? 1 : (kk > 63 ? 63 : kk);

  if (tid == 0) {
    float M = wsM[row * SPLITS];
    for (int s = 1; s < SPLITS; ++s) M = fmaxf(M, wsM[row * SPLITS + s]);
    float S1 = 0.f, St = 0.f;
    for (int s = 0; s < SPLITS; ++s) {
      float mw = wsM[row * SPLITS + s];
      S1 += wsS1[row * SPLITS + s] * __expf(mw - M);
      St += wsSt[row * SPLITS + s] * __expf((mw - M) * inv_t);
    }
    sM = M; sS1 = S1; sSt = St;
  }

  unsigned nc = wsCnt[row]; if (nc > CAP) nc = CAP;
  for (int i = tid; i < CAP; i += BLOCK) {
    if (i < (int)nc) {
      candVal[i] = wsVal[(size_t)row * CAP + i];
      candIdx[i] = wsIdx[(size_t)row * CAP + i];
    } else {
      candVal[i] = -INFINITY;
      candIdx[i] = -1;
    }
  }
  __syncthreads();

  // Bitonic sort (descending) of CAP candidates in LDS
  for (int k = 2; k <= CAP; k <<= 1) {
    for (int j = k >> 1; j > 0; j >>= 1) {
      for (int e = tid; e < CAP; e += BLOCK) {
        int p = e ^ j;
        if (p > e) {
          float ve = candVal[e], vp = candVal[p];
          bool up = ((e & k) == 0);
          if (up ? (ve < vp) : (ve > vp)) {
            candVal[e] = vp; candVal[p] = ve;
            int ti = candIdx[e]; candIdx[e] = candIdx[p]; candIdx[p] = ti;
          }
        }
      }
      __syncthreads();
    }
  }

  const float M = sM, S1 = sS1, St = sSt;
  const float lse = M + __logf(S1);

  if (tid < 63) {
    float v = candVal[tid];
    pj[tid] = (tid < (int)nc) ? __expf((v - M) * inv_t) / St : 0.0f;
  }
  __syncthreads();
  if (tid == 0) {
    float c = 0.f;
    for (int j2 = 0; j2 < 63; ++j2) { cumb[j2] = c; c += pj[j2]; }
  }
  __syncthreads();
  if (tid < 63) {
    bool keep = (tid < kk) && (cumb[tid] <= topp) && (tid < (int)nc);
    float sc = -INFINITY;
    if (keep) sc = candVal[tid] * inv_t + gumbel[(size_t)row * V + candIdx[tid]];
    score[tid] = sc;
  }
  __syncthreads();
  if (tid == 0) {
    int best = 0; float bs = score[0];
    for (int j2 = 1; j2 < 63; ++j2)
      if (score[j2] > bs) { bs = score[j2]; best = j2; }
    if (greedy) best = 0;                    // argmax == sorted position 0
    int sampled = candIdx[best];
    out_ids[row]  = sampled;
    out_idx[(size_t)row * (nl + 1)] = sampled;
    out_lp [(size_t)row * (nl + 1)] = candVal[best] - lse;
    out_rank[row] = best + 1;                // distinct values => rank == position+1
  }
  if (tid < nl) {
    out_idx[(size_t)row * (nl + 1) + 1 + tid] = candIdx[tid];
    out_lp [(size_t)row * (nl + 1) + 1 + tid] = candVal[tid] - lse;
  }
}

extern "C" void kernel_launch(void* const* d_in, const int* in_sizes, int n_in,
                              void* d_out, int out_size, void* d_ws, size_t ws_size,
                              hipStream_t stream) {
  const float* logits      = (const float*)d_in[0];
  const float* temperature = (const float*)d_in[1];
  const float* top_p       = (const float*)d_in[2];
  const int*   top_k       = (const int*)  d_in[3];
  const float* gumbel      = (const float*)d_in[4];
  (void)n_in; (void)ws_size;

  const int B  = in_sizes[1];                 // temperature has B elements
  const int V  = in_sizes[0] / B;             // 128000
  const int nl = (out_size / B - 4) / 2;      // 1 + (nl+1) + (nl+1) + 1 slots per row

  // outputs (flat 4-byte slots, concatenated in return order)
  int*   out_ids  = (int*)d_out;
  int*   out_idx  = out_ids + B;
  float* out_lp   = (float*)(out_idx + (size_t)B * (nl + 1));
  int*   out_rank = (int*)  (out_lp  + (size_t)B * (nl + 1));

  // workspace carve-up (all 4-byte aligned)
  unsigned* wsCnt = (unsigned*)d_ws;
  float*    wsM   = (float*)(wsCnt + B);
  float*    wsS1  = wsM  + (size_t)B * SPLITS;
  float*    wsSt  = wsS1 + (size_t)B * SPLITS;
  float*    wsVal = wsSt + (size_t)B * SPLITS;
  int*      wsIdx = (int*)(wsVal + (size_t)B * CAP);

  k0_init<<<(B + BLOCK - 1) / BLOCK, BLOCK, 0, stream>>>(wsCnt, B);
  k1_scan<<<B * SPLITS, BLOCK, 0, stream>>>(logits, temperature,
                                            wsCnt, wsM, wsS1, wsSt, wsVal, wsIdx, V);
  k2_finalize<<<B, BLOCK, 0, stream>>>(temperature, top_p, top_k, gumbel,
                                       wsCnt, wsM, wsS1, wsSt, wsVal, wsIdx,
                                       out_ids, out_idx, out_lp, out_rank, V, nl);
}